// PatchFreqLoss_80900003987550
// MI455X (gfx1250) — compile-verified
//
#include <hip/hip_runtime.h>
#include <math.h>

// ---------------------------------------------------------------------------
// PatchFreqLoss on MI455X (gfx1250)
//
// Math: per 16x16 patch, rfft2(pred)-rfft2(target) = rfft2(diff) (DFT linear).
// 2-D DFT of a 16x16 tile = F * X * F with F the symmetric 16x16 DFT matrix
// (F = C - i*S). Implemented as chained V_WMMA_F32_16X16X4_F32 ops:
//   Zr = C*X, Zi = Sn*X (Sn = -S), Yr = Zr*C - Zi*Sn, Yi = Zi*C + Zr*Sn.
// Keep rfft bins k2 in [0,8] (lane&15 in the C/D layout), sum |Y|^2 over
// channels, sqrt per patch, mean over 32768 patches.
//
// Data path: TDM (tensor_load_to_lds) DMAs a 256x16 f32 tile (row stride 512)
// per channel/half-strip into LDS; waves build WMMA operands from LDS.
// Bandwidth bound: 201 MB @ 23.3 TB/s ~ 8.6 us; WMMA math is negligible.
// ---------------------------------------------------------------------------

typedef float v2f __attribute__((ext_vector_type(2)));
typedef float v8f __attribute__((ext_vector_type(8)));
typedef unsigned int v4u __attribute__((ext_vector_type(4)));
typedef int v4i __attribute__((ext_vector_type(4)));
typedef int v8i __attribute__((ext_vector_type(8)));

#if defined(__has_builtin)
#if __has_builtin(__builtin_amdgcn_tensor_load_to_lds) && \
    __has_builtin(__builtin_amdgcn_s_wait_tensorcnt)
#define HAVE_TDM 1
#endif
#endif
#ifndef HAVE_TDM
#define HAVE_TDM 0
#endif

union V8U { v8f v; float f[8]; };

__device__ __forceinline__ v8f wmma4(v2f a, v2f b, v8f c) {
  // D = A(16x4) * B(4x16) + C, f32 accumulate
  return __builtin_amdgcn_wmma_f32_16x16x4_f32(false, a, false, b, (short)0, c,
                                               false, false);
}

#if HAVE_TDM
// Issue a TDM load of a 2-D f32 tile: tile_dim0=256 cols, tile_dim1=16 rows,
// tensor row stride 512 floats, dense into LDS at lds_byte_off.
// This toolchain exposes the 6-arg builtin:
//   (uint32x4 g0, int32x8 g1, int32x4 g2, int32x4 g3, int32x8 extra, i32 cpol)
// Groups 2/3 + extra group are only used by 3D+/iterate/gather modes; a 2-D
// tile (tile_dim2 = 0) leaves them zero.
__device__ __forceinline__ void tdm_load_tile(const float* gptr,
                                              unsigned int lds_byte_off) {
  unsigned long long ga = (unsigned long long)(uintptr_t)gptr;
  v4u g0;
  g0[0] = 1u;                                    // count=1, user descriptor
  g0[1] = lds_byte_off;                          // lds_addr
  g0[2] = (unsigned int)(ga & 0xffffffffull);    // global_addr[31:0]
  g0[3] = (unsigned int)((ga >> 32) & 0x01ffffffull) | (2u << 30); // type=2
  v8i g1;
  g1[0] = (int)(2u << 16);      // data_size = 4 bytes
  g1[1] = (int)(512u << 16);    // tensor_dim0 = 512 (low 16 bits)
  g1[2] = (int)(16u << 16);     // tensor_dim0 hi=0 | tensor_dim1 = 16
  g1[3] = (int)(256u << 16);    // tensor_dim1 hi=0 | tile_dim0 = 256
  g1[4] = 16;                   // tile_dim1 = 16 | tile_dim2 = 0
  g1[5] = 512;                  // tensor_dim0_stride low = 512 elements
  g1[6] = 0;                    // stride0 hi | dim1_stride lo (unused)
  g1[7] = 0;
  v4i gz4 = {0, 0, 0, 0};               // groups 2/3 unused (2-D tile)
  v8i gz8 = {0, 0, 0, 0, 0, 0, 0, 0};   // extra group unused
  __builtin_amdgcn_tensor_load_to_lds(g0, g1, gz4, gz4, gz8, 0);
}
#endif

__global__ void __launch_bounds__(256)
patch_freq_main(const float* __restrict__ pred, const float* __restrict__ target,
                float* __restrict__ patch_out) {
  __shared__ __align__(16) float lds_p[16 * 256];
  __shared__ __align__(16) float lds_t[16 * 256];
  __shared__ float lds_z[8][16 * 17];  // per-wave transpose scratch, padded

  const int tid = threadIdx.x;
  const int wid = tid >> 5;   // wave in workgroup (0..7)
  const int L = tid & 31;     // lane
  const int h = L >> 4;       // half-wave (0/1)
  const int q = L & 15;       // lane index within half

  // Twiddle fragments of symmetric matrices C(k,n)=cos(2*pi*k*n/16) and
  // Sn(k,n) = -sin(2*pi*k*n/16). Because they are symmetric, the same
  // fragment serves as A-layout (rows q, cols k0+2h..) and B-layout
  // (rows k0+2h.., col q).
  v2f Cf[4], Sf[4];
#pragma unroll
  for (int j = 0; j < 4; ++j) {
    const int k0 = 4 * j + 2 * h;
    const float a0 = (float)((q * k0) & 15) * 0.39269908169872414f;       // pi/8
    const float a1 = (float)((q * (k0 + 1)) & 15) * 0.39269908169872414f;
    Cf[j].x = __cosf(a0);
    Cf[j].y = __cosf(a1);
    Sf[j].x = -__sinf(a0);
    Sf[j].y = -__sinf(a1);
  }

  const int b = blockIdx.x >> 5;   // batch
  const int nh = blockIdx.x & 31;  // patch row block

  float acc[4] = {0.f, 0.f, 0.f, 0.f};  // per-patch |Y|^2 accum (masked lanes)

  for (int c = 0; c < 3; ++c) {
    for (int hh = 0; hh < 2; ++hh) {
      const size_t gbase =
          ((size_t)((b * 3 + c) * 512 + nh * 16)) * 512 + (size_t)hh * 256;

#if HAVE_TDM
      if (tid < 32) {  // wave 0 drives the DMA engine
        tdm_load_tile(pred + gbase, (unsigned int)(uintptr_t)&lds_p[0]);
        tdm_load_tile(target + gbase, (unsigned int)(uintptr_t)&lds_t[0]);
        __builtin_amdgcn_s_wait_tensorcnt((short)0);
      }
#else
      // Fallback: cooperative float4 copy (coalesced)
      for (int i = tid; i < 1024; i += 256) {
        const int r = i >> 6;       // row 0..15
        const int c4 = i & 63;      // float4 within 256-wide row
        ((float4*)&lds_p[r * 256])[c4] =
            ((const float4*)(pred + gbase + (size_t)r * 512))[c4];
        ((float4*)&lds_t[r * 256])[c4] =
            ((const float4*)(target + gbase + (size_t)r * 512))[c4];
      }
#endif
      __syncthreads();

      for (int s = 0; s < 2; ++s) {
        const int lp = wid + 8 * s;       // local patch 0..15 in this half
        const int col = lp * 16 + q;      // column in the 256-wide tile

        // ---- Stage 1: Zr = C*X, Zi = Sn*X (X = diff patch, B layout) ----
        v8f zr = {0, 0, 0, 0, 0, 0, 0, 0};
        v8f zi = {0, 0, 0, 0, 0, 0, 0, 0};
#pragma unroll
        for (int j = 0; j < 4; ++j) {
          const int r0 = 4 * j + 2 * h;
          v2f xb;
          xb.x = lds_p[r0 * 256 + col] - lds_t[r0 * 256 + col];
          xb.y = lds_p[(r0 + 1) * 256 + col] - lds_t[(r0 + 1) * 256 + col];
          zr = wmma4(Cf[j], xb, zr);
          zi = wmma4(Sf[j], xb, zi);
        }

        // ---- Re-lay Zr, Zi from C/D layout to A layout via LDS ----
        float* zbuf = lds_z[wid];
        V8U uz;
        uz.v = zr;
#pragma unroll
        for (int i = 0; i < 8; ++i) zbuf[(i + 8 * h) * 17 + q] = uz.f[i];
        v2f azr[4];
#pragma unroll
        for (int j = 0; j < 4; ++j) {
          const int k = 4 * j + 2 * h;
          azr[j].x = zbuf[q * 17 + k];
          azr[j].y = zbuf[q * 17 + k + 1];
        }
        uz.v = zi;  // same-wave DS ops are in-order: safe to overwrite
#pragma unroll
        for (int i = 0; i < 8; ++i) zbuf[(i + 8 * h) * 17 + q] = uz.f[i];
        v2f azi[4];
#pragma unroll
        for (int j = 0; j < 4; ++j) {
          const int k = 4 * j + 2 * h;
          azi[j].x = zbuf[q * 17 + k];
          azi[j].y = zbuf[q * 17 + k + 1];
        }

        // ---- Stage 2: Yi = Zi*C + Zr*Sn ; Yr = Zr*C - Zi*Sn ----
        v8f yi = {0, 0, 0, 0, 0, 0, 0, 0};
#pragma unroll
        for (int j = 0; j < 4; ++j) yi = wmma4(azi[j], Cf[j], yi);
#pragma unroll
        for (int j = 0; j < 4; ++j) yi = wmma4(azr[j], Sf[j], yi);
        v8f p = {0, 0, 0, 0, 0, 0, 0, 0};
#pragma unroll
        for (int j = 0; j < 4; ++j) p = wmma4(azi[j], Sf[j], p);
        v8f yr = -p;
#pragma unroll
        for (int j = 0; j < 4; ++j) yr = wmma4(azr[j], Cf[j], yr);

        // ---- |Y|^2 over rfft bins: k2 = q <= 8, all k1 ----
        V8U ur, uim;
        ur.v = yr;
        uim.v = yi;
        float part = 0.f;
#pragma unroll
        for (int i = 0; i < 8; ++i)
          part += ur.f[i] * ur.f[i] + uim.f[i] * uim.f[i];
        if (q > 8) part = 0.f;
        acc[hh * 2 + s] += part;
      }
      __syncthreads();  // done reading LDS before next DMA overwrites it
    }
  }

  // Cross-lane reduce + per-patch sqrt
#pragma unroll
  for (int idx = 0; idx < 4; ++idx) {
    const int hh = idx >> 1, s = idx & 1;
    float v = acc[idx];
#pragma unroll
    for (int off = 16; off > 0; off >>= 1) v += __shfl_xor(v, off, 32);
    if (L == 0) {
      const int nw = hh * 16 + wid + 8 * s;
      patch_out[blockIdx.x * 32 + nw] = sqrtf(v);
    }
  }
}

__global__ void __launch_bounds__(256)
patch_freq_reduce(const float* __restrict__ patch_vals, float* __restrict__ out) {
  __shared__ float red[256];
  float s = 0.f;
  for (int i = threadIdx.x; i < 32768; i += 256) s += patch_vals[i];
  red[threadIdx.x] = s;
  __syncthreads();
  for (int off = 128; off > 0; off >>= 1) {
    if (threadIdx.x < off) red[threadIdx.x] += red[threadIdx.x + off];
    __syncthreads();
  }
  if (threadIdx.x == 0) out[0] = red[0] * (1.0f / 32768.0f);
}

extern "C" void kernel_launch(void* const* d_in, const int* in_sizes, int n_in,
                              void* d_out, int out_size, void* d_ws,
                              size_t ws_size, hipStream_t stream) {
  const float* pred = (const float*)d_in[0];
  const float* target = (const float*)d_in[1];
  float* pvals = (float*)d_ws;  // 32768 floats, fully overwritten each launch

  patch_freq_main<<<1024, 256, 0, stream>>>(pred, target, pvals);
  patch_freq_reduce<<<1, 256, 0, stream>>>(pvals, (float*)d_out);
}